// ScorePerm1D_29858612641943
// MI455X (gfx1250) — compile-verified
//
#include <hip/hip_runtime.h>
#include <hip/hip_bf16.h>
#include <math.h>

#define D 8192
#define NTHREADS 1024
#define EPSF 1e-8f
#define SINK_ITERS 10

typedef float v2f __attribute__((ext_vector_type(2)));
typedef float v8f __attribute__((ext_vector_type(8)));

// ---------------------------------------------------------------------------
// Kernel 1: standardize score, bitonic sort (value,index) in LDS.
//   ws_a[k]    = k-th smallest standardized score
//   ws_perm[k] = original index of k-th smallest  (== argsort result)
// ---------------------------------------------------------------------------
__global__ __launch_bounds__(NTHREADS)
void k1_normalize_sort(const float* __restrict__ score,
                       float* __restrict__ ws_a, int* __restrict__ ws_perm)
{
    __shared__ float sv[D];   // 32 KB
    __shared__ int   si[D];   // 32 KB (reused as float scratch for reductions)
    float* red = (float*)si;

    const int t = threadIdx.x, lane = t & 31, wid = t >> 5;

    // mean / std (ddof=1) via sum and sum-of-squares
    float ps = 0.f, ps2 = 0.f;
    for (int e = t; e < D; e += NTHREADS) { float x = score[e]; ps += x; ps2 += x * x; }
    for (int off = 16; off > 0; off >>= 1) {
        ps  += __shfl_down(ps,  off);
        ps2 += __shfl_down(ps2, off);
    }
    if (lane == 0) { red[wid] = ps; red[32 + wid] = ps2; }
    __syncthreads();
    if (t == 0) {
        float S = 0.f, S2 = 0.f;
        for (int w = 0; w < 32; ++w) { S += red[w]; S2 += red[32 + w]; }
        float mean = S / (float)D;
        float var  = (S2 - (float)D * mean * mean) / (float)(D - 1);
        red[64] = mean;
        red[65] = sqrtf(fmaxf(var, 0.f)) + 1e-6f;   // std + 1e-6
    }
    __syncthreads();
    const float mean = red[64], denom = red[65];
    __syncthreads();                                 // done with red before si init

    for (int e = t; e < D; e += NTHREADS) { sv[e] = (score[e] - mean) / denom; si[e] = e; }

    // bitonic sort ascending, tiebreak on original index
    for (int k = 2; k <= D; k <<= 1) {
        for (int j = k >> 1; j > 0; j >>= 1) {
            __syncthreads();
            for (int base = 0; base < D; base += NTHREADS) {
                int idx = base + t;
                int ixj = idx ^ j;
                if (ixj > idx) {
                    float v0 = sv[idx], v1 = sv[ixj];
                    int   i0 = si[idx], i1 = si[ixj];
                    bool lt = (v0 < v1) || (v0 == v1 && i0 < i1);
                    bool up = ((idx & k) == 0);
                    if (lt != up) { sv[idx] = v1; sv[ixj] = v0; si[idx] = i1; si[ixj] = i0; }
                }
            }
        }
    }
    __syncthreads();
    for (int e = t; e < D; e += NTHREADS) { ws_a[e] = sv[e]; ws_perm[e] = si[e]; }
}

// ---------------------------------------------------------------------------
// Dual inclusive prefix scan of w1[k]*x[k] and w2[k]*x[k] over D=8192 elements
// with 1024 threads (8 contiguous elems each). wave32 shfl scans.
// ---------------------------------------------------------------------------
__device__ inline void scan_pair_8k(const float* __restrict__ w1,
                                    const float* __restrict__ w2,
                                    const float* __restrict__ x,
                                    float* __restrict__ o1,
                                    float* __restrict__ o2,
                                    float* lds1, float* lds2)
{
    const int t = threadIdx.x, lane = t & 31, wid = t >> 5;
    const int base = t * 8;
    float v1[8], v2[8];
    float s1 = 0.f, s2 = 0.f;
#pragma unroll
    for (int q = 0; q < 8; ++q) {
        float xx = x[base + q];
        s1 += w1[base + q] * xx; v1[q] = s1;
        s2 += w2[base + q] * xx; v2[q] = s2;
    }
    float t1 = s1, t2 = s2;
#pragma unroll
    for (int d = 1; d < 32; d <<= 1) {
        float y1 = __shfl_up(t1, d);
        float y2 = __shfl_up(t2, d);
        if (lane >= d) { t1 += y1; t2 += y2; }
    }
    if (lane == 31) { lds1[wid] = t1; lds2[wid] = t2; }
    __syncthreads();
    if (wid == 0) {
        float a1 = lds1[lane], a2 = lds2[lane];
#pragma unroll
        for (int d = 1; d < 32; d <<= 1) {
            float y1 = __shfl_up(a1, d);
            float y2 = __shfl_up(a2, d);
            if (lane >= d) { a1 += y1; a2 += y2; }
        }
        lds1[lane] = a1; lds2[lane] = a2;
    }
    __syncthreads();
    const float off1 = (wid ? lds1[wid - 1] : 0.f) + (t1 - s1);
    const float off2 = (wid ? lds2[wid - 1] : 0.f) + (t2 - s2);
#pragma unroll
    for (int q = 0; q < 8; ++q) { o1[base + q] = v1[q] + off1; o2[base + q] = v2[q] + off2; }
}

// ---------------------------------------------------------------------------
// Kernel 2: Sinkhorn on the diag(u) K diag(v) factorization, O(D) per half-
// iteration via the L1-kernel prefix/suffix identity in sorted order:
//   (K x)_i = e^{-a_i/t} * sum_{k<=i} e^{a_k/t} x_k
//           + e^{ a_i/t} * sum_{k> i} e^{-a_k/t} x_k
// Emits U1,U2 (sorted-row domain) and V1,V2 (original-column domain) with
//   P_soft[i,j] = min(U1_i*V1_j, U2_i*V2_j)
// Also writes perm_hard (as float) into d_out[0..D).
// ---------------------------------------------------------------------------
__global__ __launch_bounds__(NTHREADS)
void k2_sinkhorn(const float* __restrict__ tau_p,
                 const float* __restrict__ ws_a, const int* __restrict__ ws_perm,
                 float* __restrict__ epa, float* __restrict__ ema,
                 float* __restrict__ u,   float* __restrict__ vs,
                 float* __restrict__ buf1, float* __restrict__ buf2,
                 float* __restrict__ U1, float* __restrict__ U2,
                 float* __restrict__ V1, float* __restrict__ V2,
                 float* __restrict__ out_perm)
{
    __shared__ float lds1[32], lds2[32];
    const int t = threadIdx.x;
    const float tau  = fmaxf(tau_p[0], 1e-6f);
    const float itau = 1.0f / tau;

    for (int e = t; e < D; e += NTHREADS) {
        float a = ws_a[e];
        epa[e] = __expf(a * itau);
        ema[e] = __expf(-a * itau);
        u[e] = 1.f; vs[e] = 1.f;
    }
    __syncthreads();

    for (int it = 0; it < SINK_ITERS; ++it) {
        // row normalization: rowsum_i = u_i * (Kv)_i
        scan_pair_8k(epa, ema, vs, buf1, buf2, lds1, lds2);
        __syncthreads();
        float T2 = buf2[D - 1];
        for (int e = t; e < D; e += NTHREADS) {
            float rowfac = ema[e] * buf1[e] + epa[e] * (T2 - buf2[e]);
            float uo = u[e];
            u[e] = uo / (uo * rowfac + EPSF);
        }
        __syncthreads();
        // column normalization: colsum_k = vs_k * (K^T u)_k (K symmetric in sorted domain)
        scan_pair_8k(epa, ema, u, buf1, buf2, lds1, lds2);
        __syncthreads();
        T2 = buf2[D - 1];
        for (int e = t; e < D; e += NTHREADS) {
            float colfac = ema[e] * buf1[e] + epa[e] * (T2 - buf2[e]);
            float vo = vs[e];
            vs[e] = vo / (vo * colfac + EPSF);
        }
        __syncthreads();
    }

    for (int e = t; e < D; e += NTHREADS) {
        float ue = u[e], ve = vs[e];
        U1[e] = ue * ema[e];
        U2[e] = ue * epa[e];
        int p = ws_perm[e];
        V1[p] = ve * epa[e];   // scatter to original-column order
        V2[p] = ve * ema[e];
        out_perm[e] = (float)p;
    }
}

// ---------------------------------------------------------------------------
// Kernel 3: materialize P_soft / P_hard / P_ste.
// One wave per 16x16 tile. Each tile = elementwise min of two rank-1 outer
// products, computed on the matrix pipe with V_WMMA_F32_16X16X4_F32:
// A has only K=0 column (lanes 0-15, VGPR0), B only K=0 row -> D = a (x) b.
// Bandwidth-bound: ~805 MB of stores total.
// ---------------------------------------------------------------------------
__global__ __launch_bounds__(256)
void k3_tiles(const float* __restrict__ U1, const float* __restrict__ U2,
              const float* __restrict__ V1, const float* __restrict__ V2,
              const int* __restrict__ perm,
              float* __restrict__ out_soft, float* __restrict__ out_hard,
              float* __restrict__ out_ste)
{
    const int lane = threadIdx.x & 31;
    const int wave = threadIdx.x >> 5;
    const int jt = blockIdx.x * 8 + wave;   // 0..511
    const int it = blockIdx.y;              // 0..511
    const int i0 = it * 16, j0 = jt * 16;
    const bool lo = lane < 16;
    const int  l15 = lane & 15;

    v2f a1 = {0.f, 0.f}, a2 = {0.f, 0.f}, b1 = {0.f, 0.f}, b2 = {0.f, 0.f};
    if (lo) {
        a1.x = U1[i0 + lane]; a2.x = U2[i0 + lane];   // A: M=lane, K=0
        b1.x = V1[j0 + lane]; b2.x = V2[j0 + lane];   // B: K=0, N=lane
    }

    v8f z = {};
    v8f d1 = __builtin_amdgcn_wmma_f32_16x16x4_f32(
        false, a1, false, b1, (short)0, z, false, false);
    v8f d2 = __builtin_amdgcn_wmma_f32_16x16x4_f32(
        false, a2, false, b2, (short)0, z, false, false);

    const int mbase = i0 + (lo ? 0 : 8);    // C/D layout: VGPR r -> M = r (+8 for hi half)
    const int j = j0 + l15;
#pragma unroll
    for (int r = 0; r < 8; ++r) {
        int i = mbase + r;
        float p = fminf(d1[r], d2[r]);
        float h = (perm[i] == j) ? 1.0f : 0.0f;
        size_t off = (size_t)i * (size_t)D + (size_t)j;
        out_soft[off] = p;
        out_hard[off] = h;
        out_ste[off]  = (h - p) + p;        // matches P_hard - sg(P_soft) + P_soft
    }
}

// ---------------------------------------------------------------------------
extern "C" void kernel_launch(void* const* d_in, const int* in_sizes, int n_in,
                              void* d_out, int out_size, void* d_ws, size_t ws_size,
                              hipStream_t stream)
{
    const float* score = (const float*)d_in[0];
    const float* tau   = (const float*)d_in[1];
    float* out = (float*)d_out;

    float* ws      = (float*)d_ws;           // 12 * D * 4B = 384 KB of scratch
    float* ws_a    = ws + 0 * D;
    int*   ws_perm = (int*)(ws + 1 * D);
    float* epa     = ws + 2 * D;
    float* ema     = ws + 3 * D;
    float* u       = ws + 4 * D;
    float* vs      = ws + 5 * D;
    float* buf1    = ws + 6 * D;
    float* buf2    = ws + 7 * D;
    float* U1      = ws + 8 * D;
    float* U2      = ws + 9 * D;
    float* V1      = ws + 10 * D;
    float* V2      = ws + 11 * D;

    float* out_perm = out;                               // D floats
    float* out_soft = out + D;                           // D*D
    float* out_hard = out_soft + (size_t)D * (size_t)D;  // D*D
    float* out_ste  = out_hard + (size_t)D * (size_t)D;  // D*D

    k1_normalize_sort<<<1, NTHREADS, 0, stream>>>(score, ws_a, ws_perm);
    k2_sinkhorn<<<1, NTHREADS, 0, stream>>>(tau, ws_a, ws_perm, epa, ema, u, vs,
                                            buf1, buf2, U1, U2, V1, V2, out_perm);
    dim3 grid(D / 16 / 8, D / 16, 1);   // (64, 512): 8 waves/block, 1 tile/wave
    k3_tiles<<<grid, 256, 0, stream>>>(U1, U2, V1, V2, ws_perm,
                                       out_soft, out_hard, out_ste);
}